// RNN_54597624267294
// MI455X (gfx1250) — compile-verified
//
#include <hip/hip_runtime.h>

typedef float v2f __attribute__((ext_vector_type(2)));
typedef float v8f __attribute__((ext_vector_type(8)));

namespace {
constexpr int Bsz = 4096, Tsz = 512, Isz = 35, Hsz = 12, Osz = 35;
constexpr int SX  = 9;    // ceil(35/4) K-steps for x-projection (K padded to 36)
constexpr int SH  = 3;    // 12/4 K-steps for h-projection
constexpr int NT  = 4;    // gate N-tiles (i,f,g,o), each 16 wide (12 valid)
constexpr int NTO = 3;    // output N-tiles (48 >= 35)
constexpr int HSTR = 20;  // transpose-buffer row stride: conflict-free
constexpr int CT  = 8;    // timesteps per async chunk
constexpr int NC  = Tsz / CT;          // 64 chunks
constexpr int XSTR = 324;              // LDS row stride (floats): 324*4 % 16 == 0,
                                       // 324 % 64 == 4 -> conflict-free ds reads
// workspace layout (float2 units for B-fragments, then floats)
constexpr int F2_BX  = 0;                  // SX*NT*32  = 1152 float2
constexpr int F2_BH  = F2_BX + SX*NT*32;
constexpr int F2_BFC = F2_BH + SH*NT*32;
constexpr int F2_END = F2_BFC + SH*NTO*32;
constexpr int F_BIAS = F2_END * 2;
constexpr int F_BFC  = F_BIAS + NT*16;
constexpr int WS_FLOATS = F_BFC + NTO*16;
}

// ---------- fast activations (v_exp_f32 + v_rcp_f32) ----------
__device__ __forceinline__ float sigmoidf_fast(float x) {
  return __builtin_amdgcn_rcpf(1.0f + __expf(-x));
}
__device__ __forceinline__ float tanhf_fast(float x) {
  return 2.0f * __builtin_amdgcn_rcpf(1.0f + __expf(-2.0f * x)) - 1.0f;
}

__device__ __forceinline__ v8f wmma4(v2f a, v2f b, v8f c) {
  // D = A(16x4 f32) * B(4x16 f32) + C(16x16 f32)
  return __builtin_amdgcn_wmma_f32_16x16x4_f32(false, a, false, b, (short)0, c,
                                               false, false);
}

// CDNA5 async global->LDS copy (ASYNCcnt-tracked, no VGPR round trip).
// l must be 16B-aligned; byteoff is a compile-time immediate.
__device__ __forceinline__ void async_ld_b128(const float* g, float* l, int byteoff) {
  unsigned ldsoff = (unsigned)(size_t)l;  // low 32 bits of LDS-aperture address
  asm volatile("global_load_async_to_lds_b128 %0, %1, off offset:%2"
               :: "v"(ldsoff), "v"(g), "n"(byteoff) : "memory");
}
#define WAIT_ASYNC_LE35() asm volatile("s_wait_asynccnt 0x23" ::: "memory")
#define WAIT_DS0()        asm volatile("s_wait_dscnt 0x0" ::: "memory")

// ============================================================================
// Pack kernel: swizzle tiny weights into WMMA B-fragment layout in workspace.
// B-frag layout (4x16 f32, 2 VGPRs/lane): lane l, component cpt:
//   k = 4*s + 2*(l>>4) + cpt,  n = 16*tile + (l&15)
// ============================================================================
__global__ void lstm_pack(const float* __restrict__ w_ih, const float* __restrict__ w_hh,
                          const float* __restrict__ b_ih, const float* __restrict__ b_hh,
                          const float* __restrict__ w_fc, const float* __restrict__ b_fc,
                          float* __restrict__ ws) {
  const int tid = threadIdx.x + blockIdx.x * blockDim.x;
  const int nth = blockDim.x * gridDim.x;

  for (int idx = tid; idx < SX * NT * 32; idx += nth) {
    int lane = idx & 31, gt = (idx >> 5) & 3, s = idx >> 7;
    int unit = lane & 15;
    int kb = 4 * s + ((lane >> 4) << 1);
    float v0 = 0.f, v1 = 0.f;
    if (unit < Hsz) {
      int row = gt * Hsz + unit;
      if (kb     < Isz) v0 = w_ih[row * Isz + kb];
      if (kb + 1 < Isz) v1 = w_ih[row * Isz + kb + 1];
    }
    ws[2 * (F2_BX + idx)]     = v0;
    ws[2 * (F2_BX + idx) + 1] = v1;
  }
  for (int idx = tid; idx < SH * NT * 32; idx += nth) {
    int lane = idx & 31, gt = (idx >> 5) & 3, s = idx >> 7;
    int unit = lane & 15;
    int kb = 4 * s + ((lane >> 4) << 1);
    float v0 = 0.f, v1 = 0.f;
    if (unit < Hsz) {
      int row = gt * Hsz + unit;
      v0 = w_hh[row * Hsz + kb];
      v1 = w_hh[row * Hsz + kb + 1];
    }
    ws[2 * (F2_BH + idx)]     = v0;
    ws[2 * (F2_BH + idx) + 1] = v1;
  }
  for (int idx = tid; idx < SH * NTO * 32; idx += nth) {
    int lane = idx & 31, tn = (idx >> 5) % NTO, s = idx / (32 * NTO);
    int n = tn * 16 + (lane & 15);
    int kb = 4 * s + ((lane >> 4) << 1);
    float v0 = 0.f, v1 = 0.f;
    if (n < Osz) {
      v0 = w_fc[n * Hsz + kb];
      v1 = w_fc[n * Hsz + kb + 1];
    }
    ws[2 * (F2_BFC + idx)]     = v0;
    ws[2 * (F2_BFC + idx) + 1] = v1;
  }
  for (int n = tid; n < NT * 16; n += nth) {
    int unit = n & 15, gt = n >> 4;
    ws[F_BIAS + n] = (unit < Hsz) ? (b_ih[gt * Hsz + unit] + b_hh[gt * Hsz + unit]) : 0.f;
  }
  for (int n = tid; n < NTO * 16; n += nth)
    ws[F_BFC + n] = (n < Osz) ? b_fc[n] : 0.f;
}

// ============================================================================
// Fused LSTM: one wave (= one workgroup) owns 16 batch rows for all 512 steps.
//  - x staged in 8-timestep chunks via async global->LDS b128, double buffered
//  - recurrent h transposed C-layout -> A-layout via tiny LDS round trip
//  - ping-pong accumulators: x-projection of step t+1 overlaps the recurrent
//    WMMA chain + activations of step t
//  - persistent bias registers feed the first WMMA as its C operand
// ============================================================================
__global__ __launch_bounds__(32) void lstm_fused(const float* __restrict__ x,
                                                 const float* __restrict__ ws,
                                                 float* __restrict__ out) {
  __shared__ float xbuf[2][16 * XSTR];   // 2 x 20736 B async staging buffers
  __shared__ float hbuf[16 * HSTR];      // transpose scratch

  const int lane  = threadIdx.x & 31;
  const int r0    = blockIdx.x * 16;     // batch-row base of this wave
  const int nlo   = lane & 15;           // C-layout column / A-layout row
  const int hi    = lane >> 4;
  const int mbase = hi << 3;             // C-layout: m = j + mbase
  const int khalf = hi << 1;             // A-layout: k = 4s + khalf + {0,1}

  // zero the single pad word per row that the K-tail A-frag read can touch
  // (col CT*Isz = 280; async writes never touch cols >= 280, so it stays 0)
  xbuf[lane >> 4][(lane & 15) * XSTR + CT * Isz] = 0.f;

  // --- load all B-fragments into VGPRs (resident for whole kernel) ---
  const v2f* wsf2 = (const v2f*)ws;
  v2f bx[SX][NT];
#pragma unroll
  for (int s = 0; s < SX; ++s)
#pragma unroll
    for (int g = 0; g < NT; ++g)
      bx[s][g] = wsf2[F2_BX + (s * NT + g) * 32 + lane];
  v2f bh[SH][NT];
#pragma unroll
  for (int s = 0; s < SH; ++s)
#pragma unroll
    for (int g = 0; g < NT; ++g)
      bh[s][g] = wsf2[F2_BH + (s * NT + g) * 32 + lane];

  // persistent bias accumulator seeds (C operand of first x-proj WMMA)
  v8f biasv[NT];
#pragma unroll
  for (int g = 0; g < NT; ++g) {
    float bb = ws[F_BIAS + g * 16 + nlo];
#pragma unroll
    for (int j = 0; j < 8; ++j) biasv[g][j] = bb;
  }

  // --- async chunk copy: lane = (row, half), 35 x b128 covers 16 rows x 1120B
  const int arow  = lane >> 1;
  const int ahalf = lane & 1;
  const float* gsrc0 = x + (size_t)(r0 + arow) * Tsz * Isz + ahalf * 140;
  auto issue_chunk = [&](int tchunk, float* cb) {
    const float* gp = gsrc0 + (size_t)tchunk * (CT * Isz);
    float* lp = cb + arow * XSTR + ahalf * 140;
#pragma unroll
    for (int i = 0; i < 35; ++i) async_ld_b128(gp, lp, i * 16);
  };

  // x-projection of local step tl of resident chunk -> acc (bias + x @ w_ih^T)
  // K padded to 36; weight col K=35 is zero, so unconditional reads are exact.
  auto xproj = [&](const float* cb, int tl, v8f (&acc)[NT]) {
    const float* xr = cb + nlo * XSTR + tl * Isz;
    {
      v2f a;
      a.x = xr[khalf];
      a.y = xr[khalf + 1];
#pragma unroll
      for (int g = 0; g < NT; ++g) acc[g] = wmma4(a, bx[0][g], biasv[g]);
    }
#pragma unroll
    for (int s = 1; s < SX; ++s) {
      int k = 4 * s + khalf;
      v2f a;
      a.x = xr[k];
      a.y = xr[k + 1];
#pragma unroll
      for (int g = 0; g < NT; ++g) acc[g] = wmma4(a, bx[s][g], acc[g]);
    }
  };

  v8f h{}, c{};

  // one LSTM step: cur already holds bias + x-proj(t); adds h-proj, then
  // (overlapped) builds bias + x-proj(t+1) into nxt, then activations.
  auto step = [&](const float* cb, int tl, v8f (&cur)[NT], v8f (&nxt)[NT],
                  bool doNext) {
#pragma unroll
    for (int j = 0; j < 8; ++j) hbuf[(mbase + j) * HSTR + nlo] = h[j];
    WAIT_DS0();
#pragma unroll
    for (int s = 0; s < SH; ++s) {
      int k = 4 * s + khalf;
      v2f a;
      a.x = hbuf[nlo * HSTR + k];
      a.y = hbuf[nlo * HSTR + k + 1];
#pragma unroll
      for (int g = 0; g < NT; ++g) cur[g] = wmma4(a, bh[s][g], cur[g]);
    }
    if (doNext) xproj(cb, tl + 1, nxt);
#pragma unroll
    for (int j = 0; j < 8; ++j) {
      float ig = sigmoidf_fast(cur[0][j]);
      float fg = sigmoidf_fast(cur[1][j]);
      float gg = tanhf_fast(cur[2][j]);
      float og = sigmoidf_fast(cur[3][j]);
      float cn = fg * c[j] + ig * gg;
      c[j] = cn;
      h[j] = og * tanhf_fast(cn);
    }
  };

  // prologue: two chunks in flight
  issue_chunk(0, xbuf[0]);
  issue_chunk(1, xbuf[1]);

  v8f accA[NT], accB[NT];
  for (int tc = 0; tc < NC; ++tc) {
    float* cb = xbuf[tc & 1];
    WAIT_ASYNC_LE35();            // in-order completion => chunk tc resident
    xproj(cb, 0, accA);           // first step of chunk
    for (int tl = 0; tl < CT; tl += 2) {
      step(cb, tl,     accA, accB, true);
      step(cb, tl + 1, accB, accA, tl + 2 < CT);
    }
    WAIT_DS0();                   // ds reads of cb done before overwrite
    int tn = (tc + 2 < NC) ? (tc + 2) : (NC - 1);  // clamp keeps wait invariant
    issue_chunk(tn, cb);
  }

  // ---- epilogue: out = relu(h_last) @ w_fc^T + b_fc, via WMMA ----
#pragma unroll
  for (int j = 0; j < 8; ++j)
    hbuf[(mbase + j) * HSTR + nlo] = fmaxf(h[j], 0.f);
  WAIT_DS0();

  v2f bo[SH][NTO];
#pragma unroll
  for (int s = 0; s < SH; ++s)
#pragma unroll
    for (int tn = 0; tn < NTO; ++tn)
      bo[s][tn] = wsf2[F2_BFC + (s * NTO + tn) * 32 + lane];

  v8f obias[NTO];
#pragma unroll
  for (int tn = 0; tn < NTO; ++tn) {
    float bb = ws[F_BFC + tn * 16 + nlo];
#pragma unroll
    for (int j = 0; j < 8; ++j) obias[tn][j] = bb;
  }
  v8f oacc[NTO];
#pragma unroll
  for (int s = 0; s < SH; ++s) {
    int k = 4 * s + khalf;
    v2f a;
    a.x = hbuf[nlo * HSTR + k];
    a.y = hbuf[nlo * HSTR + k + 1];
#pragma unroll
    for (int tn = 0; tn < NTO; ++tn)
      oacc[tn] = wmma4(a, bo[s][tn], (s == 0) ? obias[tn] : oacc[tn]);
  }
#pragma unroll
  for (int tn = 0; tn < NTO; ++tn) {
    int n = tn * 16 + nlo;
    if (n < Osz) {
#pragma unroll
      for (int j = 0; j < 8; ++j)
        out[(size_t)(r0 + mbase + j) * Osz + n] = oacc[tn][j];
    }
  }
}

extern "C" void kernel_launch(void* const* d_in, const int* in_sizes, int n_in,
                              void* d_out, int out_size, void* d_ws, size_t ws_size,
                              hipStream_t stream) {
  const float* x    = (const float*)d_in[0];
  const float* w_ih = (const float*)d_in[1];
  const float* w_hh = (const float*)d_in[2];
  const float* b_ih = (const float*)d_in[3];
  const float* b_hh = (const float*)d_in[4];
  const float* w_fc = (const float*)d_in[5];
  const float* b_fc = (const float*)d_in[6];
  float* out = (float*)d_out;
  float* ws  = (float*)d_ws;
  if (ws_size < (size_t)WS_FLOATS * sizeof(float)) return;

  lstm_pack<<<2, 256, 0, stream>>>(w_ih, w_hh, b_ih, b_hh, w_fc, b_fc, ws);
  // 4096 rows / 16 per wave = 256 single-wave workgroups
  lstm_fused<<<Bsz / 16, 32, 0, stream>>>(x, ws, out);
}